// Intra_Attention_68547678044689
// MI455X (gfx1250) — compile-verified
//
#include <hip/hip_runtime.h>

typedef float v2f __attribute__((ext_vector_type(2)));
typedef float v8f __attribute__((ext_vector_type(8)));

#define BDIM  2
#define NDIM  1024
#define CDIM  64
#define SCALE 0.25f              // 1/sqrt(d), d = 16
#define LOG2E 1.44269504088896340736f

// One block per (batch*channel, i-block of 256 rows). 256 threads = 8 waves.
// Each wave owns 2 row-tiles of 16. The N x N per-channel softmax-matvec is
// driven through V_WMMA_F32_16X16X4_F32: A = 16x4 tile of exp(q_i*ks_j - m_i),
// B = 4x16 with col0 = v[j..j+3], col1 = ones -> D col0 = numerator,
// col1 = denominator, accumulated in f32 over all 1024 j.
__global__ __launch_bounds__(256)
void intra_attn_wmma_kernel(const float* __restrict__ q,
                            const float* __restrict__ k,
                            const float* __restrict__ v,
                            float* __restrict__ out) {
    __shared__ __align__(16) float ks[NDIM];   // k * (scale*log2e)
    __shared__ __align__(16) float vs[NDIM];
    __shared__ float red[256];

    const int tid  = threadIdx.x;
    const int lane = tid & 31;
    const int wave = tid >> 5;

    const int bc    = blockIdx.x;       // 0..B*C-1
    const int b     = bc >> 6;
    const int c     = bc & (CDIM - 1);
    const int ibase = blockIdx.y * 256;

    const float* qp = q   + (size_t)b * NDIM * CDIM + c;
    const float* kp = k   + (size_t)b * NDIM * CDIM + c;
    const float* vp = v   + (size_t)b * NDIM * CDIM + c;
    float*       op = out + (size_t)b * NDIM * CDIM + c;

    // Stage per-channel ks, v into LDS; track max|ks| for the safety offset.
    float lmax = 0.0f;
    for (int j = tid; j < NDIM; j += 256) {
        const float kv = kp[(size_t)j * CDIM] * (SCALE * LOG2E);
        ks[j] = kv;
        vs[j] = vp[(size_t)j * CDIM];
        lmax = fmaxf(lmax, fabsf(kv));
    }
    red[tid] = lmax;
    __syncthreads();
    for (int s = 128; s > 0; s >>= 1) {
        if (tid < s) red[tid] = fmaxf(red[tid], red[tid + s]);
        __syncthreads();
    }
    const float ksabs = red[0];

    // WMMA A-layout roles for this lane.
    const int   M    = lane & 15;             // matrix row handled by this lane
    const int   koff = (lane >> 4) << 1;      // K offset: 0 (lanes 0-15) / 2 (16-31)
    const float bc1  = (M == 1) ? 1.0f : 0.0f;
    const bool  isn0 = (M == 0);

    for (int tt = 0; tt < 2; ++tt) {
        const int   i0 = ibase + (wave * 2 + tt) * 16;
        const float a  = qp[(size_t)(i0 + M) * CDIM];   // q scalar for row M
        const float m2 = fabsf(a) * ksabs;              // >= max_j a*ks_j (log2 units)

        v8f acc0 = {};
        v8f acc1 = {};
        for (int j = 0; j < NDIM; j += 8) {
            const float2 k2a = *reinterpret_cast<const float2*>(&ks[j + koff]);
            const float2 v2a = *reinterpret_cast<const float2*>(&vs[j + koff]);
            const float2 k2b = *reinterpret_cast<const float2*>(&ks[j + 4 + koff]);
            const float2 v2b = *reinterpret_cast<const float2*>(&vs[j + 4 + koff]);

            v2f A0, B0, A1, B1;
            A0.x = __builtin_amdgcn_exp2f(__builtin_fmaf(a, k2a.x, -m2));
            A0.y = __builtin_amdgcn_exp2f(__builtin_fmaf(a, k2a.y, -m2));
            A1.x = __builtin_amdgcn_exp2f(__builtin_fmaf(a, k2b.x, -m2));
            A1.y = __builtin_amdgcn_exp2f(__builtin_fmaf(a, k2b.y, -m2));
            B0.x = isn0 ? v2a.x : bc1;
            B0.y = isn0 ? v2a.y : bc1;
            B1.x = isn0 ? v2b.x : bc1;
            B1.y = isn0 ? v2b.y : bc1;

            acc0 = __builtin_amdgcn_wmma_f32_16x16x4_f32(false, A0, false, B0,
                                                         (short)0, acc0, false, false);
            acc1 = __builtin_amdgcn_wmma_f32_16x16x4_f32(false, A1, false, B1,
                                                         (short)0, acc1, false, false);
        }
        v8f acc = acc0 + acc1;

        // D layout: VGPR r -> row r (lanes 0-15, N=lane) and row 8+r (lanes 16-31).
        // Numerator at N=0, denominator at N=1.
        #pragma unroll
        for (int r = 0; r < 8; ++r) {
            const float tlo = __shfl(acc[r], 0, 32);
            const float slo = __shfl(acc[r], 1, 32);
            const float thi = __shfl(acc[r], 16, 32);
            const float shi = __shfl(acc[r], 17, 32);
            if (lane == r)      op[(size_t)(i0 + r) * CDIM]     = tlo / slo;
            if (lane == 16 + r) op[(size_t)(i0 + 8 + r) * CDIM] = thi / shi;
        }
    }
}

extern "C" void kernel_launch(void* const* d_in, const int* in_sizes, int n_in,
                              void* d_out, int out_size, void* d_ws, size_t ws_size,
                              hipStream_t stream) {
    const float* q = (const float*)d_in[0];
    const float* k = (const float*)d_in[1];
    const float* v = (const float*)d_in[2];
    float* out = (float*)d_out;
    dim3 grid(BDIM * CDIM, NDIM / 256);
    intra_attn_wmma_kernel<<<grid, dim3(256), 0, stream>>>(q, k, v, out);
}